// WienerAttention_78030965833995
// MI455X (gfx1250) — compile-verified
//
#include <hip/hip_runtime.h>
#include <hip/hip_bf16.h>
#include <math.h>

#define FDIM 160
#define TDIM 1000
#define KORD 4

typedef __attribute__((ext_vector_type(2))) float v2f;
typedef __attribute__((ext_vector_type(8))) float v8f;

__device__ __forceinline__ float sigmoidf_(float x) {
    return 1.0f / (1.0f + __expf(-x));
}

// ---------------------------------------------------------------------------
// Kernel 1: fused query-GEMM (WMMA f32 16x16x4) + layernorm + key path +
// softmax attention weights. One wave (32 threads) handles 16 GEMM rows =
// 4 consecutive t values x 4 taps for one (b, c).
// Writes W[(b*2+c)*1000 + t][4*4] to workspace.
// ---------------------------------------------------------------------------
__global__ __launch_bounds__(32) void wiener_qkw_kernel(
    const float* __restrict__ far,     // (8,2,160,1000)
    const float* __restrict__ mix,     // (8,2,160,1000)
    const float* __restrict__ kconv_w, // (8,2)
    const float* __restrict__ kconv_b, // (8,)
    const float* __restrict__ ql_w,    // (160,160)
    const float* __restrict__ ql_b,    // (160,)
    const float* __restrict__ qn_g,
    const float* __restrict__ qn_b,
    const float* __restrict__ kl_w,    // (4,4)
    const float* __restrict__ kl_b,
    const float* __restrict__ kn_g,
    const float* __restrict__ kn_b,
    const float* __restrict__ q_vec,   // (160,)
    const float* __restrict__ k_vec,   // (4,)
    float* __restrict__ W)             // (16,1000,16) attention weights
{
    __shared__ float ldsA[16 * 161];   // staged query block (padded stride)
    __shared__ float ldsQ[16 * FDIM];  // GEMM result -> normalized q

    const int lane  = threadIdx.x;
    const int blk   = blockIdx.x;       // 0..3999
    const int bc    = blk / 250;        // b*2 + c
    const int tbase = (blk % 250) * 4;
    const int b     = bc >> 1;
    const int c     = bc & 1;
    const float sgn = c ? -1.0f : 1.0f; // conjugate: channel 1 negated

    const float* farbc = far + (size_t)bc * FDIM * TDIM;

    // ---- Stage A[row][f'] = sgn * far[b,c,f', tbase + row/4 + row%4 - 3] ----
    for (int idx = lane; idx < 16 * FDIM; idx += 32) {
        int row = idx / FDIM, fp = idx % FDIM;
        int tt = tbase + (row >> 2) + (row & 3) - (KORD - 1);
        ldsA[row * 161 + fp] = (tt >= 0) ? sgn * farbc[fp * TDIM + tt] : 0.0f;
    }
    __syncthreads();

    const int laneM = lane & 15;
    const int laneH = lane >> 4;

    // ---- GEMM: Q[row][n] = sum_k A[row][k] * ql_w[n][k] + ql_b[n] ----
    for (int nt = 0; nt < 10; ++nt) {
        const int nbase = nt * 16;
        const int ncol  = nbase + laneM;
        v8f acc = {};
        for (int ks = 0; ks < 40; ++ks) {
            const int kk = ks * 4 + laneH * 2;
            v2f a, bf;
            a.x  = ldsA[laneM * 161 + kk];
            a.y  = ldsA[laneM * 161 + kk + 1];
            bf.x = ql_w[ncol * FDIM + kk];       // B[kk][n] = ql_w[n][kk]
            bf.y = ql_w[ncol * FDIM + kk + 1];
            acc = __builtin_amdgcn_wmma_f32_16x16x4_f32(
                false, a, false, bf, (short)0, acc, false, false);
        }
        const float bias = ql_b[ncol];
        #pragma unroll
        for (int v = 0; v < 8; ++v) {
            int m = laneH ? (v + 8) : v;         // C/D layout: lanes16-31 -> M+8
            ldsQ[m * FDIM + ncol] = acc[v] + bias;
        }
    }
    __syncthreads();

    // ---- per-lane gate/affine constants over f = lane + 32*j ----
    float qsig[5], qg[5], qb[5];
    #pragma unroll
    for (int j = 0; j < 5; ++j) {
        int f = lane + 32 * j;
        qsig[j] = sigmoidf_(q_vec[f]);
        qg[j]   = qn_g[f];
        qb[j]   = qn_b[f];
    }

    // ---- layernorm over f (160) per row, then gate by sigmoid(q_vec) ----
    for (int row = 0; row < 16; ++row) {
        float s = 0.0f, s2 = 0.0f;
        #pragma unroll
        for (int j = 0; j < 5; ++j) {
            float x = ldsQ[row * FDIM + lane + 32 * j];
            s += x; s2 += x * x;
        }
        for (int off = 16; off > 0; off >>= 1) {
            s  += __shfl_xor(s,  off, 32);
            s2 += __shfl_xor(s2, off, 32);
        }
        float mean = s * (1.0f / FDIM);
        float var  = s2 * (1.0f / FDIM) - mean * mean;
        float inv  = rsqrtf(var + 1e-5f);
        #pragma unroll
        for (int j = 0; j < 5; ++j) {
            int f = lane + 32 * j;
            float x = ldsQ[row * FDIM + f];
            ldsQ[row * FDIM + f] = ((x - mean) * inv * qg[j] + qb[j]) * qsig[j];
        }
    }
    __syncthreads();

    // ---- key path constants ----
    float sk[4], klb[4], kng[4], knb[4], klw[16], kcw[8], kcb4[4];
    #pragma unroll
    for (int j = 0; j < 4; ++j) {
        sk[j]  = sigmoidf_(k_vec[j]);
        klb[j] = kl_b[j]; kng[j] = kn_g[j]; knb[j] = kn_b[j];
        int o = c * 4 + j;
        kcw[2*j]   = kconv_w[o * 2 + 0];
        kcw[2*j+1] = kconv_w[o * 2 + 1];
        kcb4[j]    = kconv_b[o];
    }
    #pragma unroll
    for (int j = 0; j < 16; ++j) klw[j] = kl_w[j];

    const float* mix0p = mix + ((size_t)(b * 2 + 0) * FDIM) * TDIM;
    const float* mix1p = mix + ((size_t)(b * 2 + 1) * FDIM) * TDIM;

    // ---- per-t: key layernorm + scores + softmax ----
    for (int tt = 0; tt < 4; ++tt) {
        const int t = tbase + tt;
        float sc[4][4];
        #pragma unroll
        for (int i = 0; i < 4; ++i)
            #pragma unroll
            for (int j = 0; j < 4; ++j) sc[i][j] = 0.0f;

        #pragma unroll
        for (int jj = 0; jj < 5; ++jj) {
            const int f = lane + 32 * jj;
            float m0 = mix0p[f * TDIM + t];
            float m1 = mix1p[f * TDIM + t];
            float kraw[4], klin[4];
            #pragma unroll
            for (int j = 0; j < 4; ++j)
                kraw[j] = kcw[2*j] * m0 + kcw[2*j+1] * m1 + kcb4[j];
            #pragma unroll
            for (int i2 = 0; i2 < 4; ++i2) {
                float a = klb[i2];
                #pragma unroll
                for (int j = 0; j < 4; ++j) a += klw[i2 * 4 + j] * kraw[j];
                klin[i2] = a;
            }
            float mn = 0.25f * (klin[0] + klin[1] + klin[2] + klin[3]);
            float vv = 0.0f;
            #pragma unroll
            for (int i2 = 0; i2 < 4; ++i2) { float d = klin[i2] - mn; vv += d * d; }
            float inv = rsqrtf(vv * 0.25f + 1e-5f);
            float kln[4];
            #pragma unroll
            for (int i2 = 0; i2 < 4; ++i2)
                kln[i2] = ((klin[i2] - mn) * inv * kng[i2] + knb[i2]) * sk[i2];
            #pragma unroll
            for (int i = 0; i < 4; ++i) {
                float q = ldsQ[(tt * 4 + i) * FDIM + f];
                #pragma unroll
                for (int i2 = 0; i2 < 4; ++i2) sc[i][i2] += q * kln[i2];
            }
        }
        for (int off = 16; off > 0; off >>= 1)
            #pragma unroll
            for (int i = 0; i < 4; ++i)
                #pragma unroll
                for (int j = 0; j < 4; ++j)
                    sc[i][j] += __shfl_xor(sc[i][j], off, 32);

        if (lane < 16) {
            int i = lane >> 2, i2 = lane & 3;
            float r0 = sc[i][0] * 0.5f, r1 = sc[i][1] * 0.5f;  // / sqrt(k)
            float r2 = sc[i][2] * 0.5f, r3 = sc[i][3] * 0.5f;
            float mx = fmaxf(fmaxf(r0, r1), fmaxf(r2, r3));
            float e0 = __expf(r0 - mx), e1 = __expf(r1 - mx);
            float e2 = __expf(r2 - mx), e3 = __expf(r3 - mx);
            float den = e0 + e1 + e2 + e3;
            float mine = (i2 == 0) ? e0 : (i2 == 1) ? e1 : (i2 == 2) ? e2 : e3;
            W[((size_t)bc * TDIM + t) * 16 + lane] = mine / den;
        }
    }
}

// ---------------------------------------------------------------------------
// Kernel 2: per (b,f,t) rank-1 complex 4x4 system build + Gauss-Jordan solve.
// ---------------------------------------------------------------------------
__global__ __launch_bounds__(256) void wiener_solve_kernel(
    const float* __restrict__ far,
    const float* __restrict__ mix,
    const float* __restrict__ v_vec,
    const float* __restrict__ W,
    float* __restrict__ out)          // (8,2,160,1000)
{
    __shared__ float lW[2][256 * 16]; // 32 KB: weights for 256 t's, both c
    const int tid   = threadIdx.x;
    const int blk   = blockIdx.x;     // 8*160*4
    const int chunk = blk & 3;
    const int f     = (blk >> 2) % FDIM;
    const int b     = blk / (FDIM * 4);
    const int t0    = chunk * 256;

    for (int cc = 0; cc < 2; ++cc) {
        const float* src = W + ((size_t)(b * 2 + cc) * TDIM + t0) * 16;
        const int limit = (TDIM - t0) * 16;
        for (int idx = tid; idx < 256 * 16; idx += 256)
            lW[cc][idx] = (idx < limit) ? src[idx] : 0.0f;
    }
    __syncthreads();

    const int t = t0 + tid;
    if (t >= TDIM) return;

    float sv[4];
    #pragma unroll
    for (int j = 0; j < 4; ++j) sv[j] = sigmoidf_(v_vec[j]);

    const float* fr = far + ((size_t)(b * 2 + 0) * FDIM + f) * TDIM;
    const float* fi = far + ((size_t)(b * 2 + 1) * FDIM + f) * TDIM;
    float ur[4], ui[4];
    #pragma unroll
    for (int i = 0; i < 4; ++i) {
        int tt = t + i - (KORD - 1);
        ur[i] = (tt >= 0) ? fr[tt] : 0.0f;
        ui[i] = (tt >= 0) ? fi[tt] : 0.0f;
    }
    float mx0 = mix[((size_t)(b * 2 + 0) * FDIM + f) * TDIM + t];
    float mx1 = mix[((size_t)(b * 2 + 1) * FDIM + f) * TDIM + t];

    const float* W0 = &lW[0][tid * 16];
    const float* W1 = &lW[1][tid * 16];

    // s_c[m] = sum_j u_c[j] * sigmoid(v)[j] * W_c[j*4+m]  (u1 = -imag)
    float s0[4], s1[4];
    #pragma unroll
    for (int m = 0; m < 4; ++m) {
        float a0 = 0.0f, a1 = 0.0f;
        #pragma unroll
        for (int j = 0; j < 4; ++j) {
            a0 += ur[j] * sv[j] * W0[j * 4 + m];
            a1 += (-ui[j]) * sv[j] * W1[j * 4 + m];
        }
        s0[m] = a0; s1[m] = a1;
    }

    // A = (1+i)(eps*ones + I) + s0 u0^T + i s1 u1^T ; y = mx0*s0 + i mx1*s1
    const float eps = 1e-8f;
    float Ar[4][4], Ai[4][4], yr[4], yi[4];
    #pragma unroll
    for (int m = 0; m < 4; ++m) {
        #pragma unroll
        for (int p = 0; p < 4; ++p) {
            float e = (m == p) ? (1.0f + eps) : eps;
            Ar[m][p] = ur[p] * s0[m] + e;
            Ai[m][p] = (-ui[p]) * s1[m] + e;
        }
        yr[m] = mx0 * s0[m];
        yi[m] = mx1 * s1[m];
    }

    // Gauss-Jordan, complex, no pivoting (identity-regularized system)
    #pragma unroll
    for (int col = 0; col < 4; ++col) {
        float dr = Ar[col][col], di = Ai[col][col];
        float inv = 1.0f / (dr * dr + di * di);
        float pr = dr * inv, pi = -di * inv;   // 1/pivot
        #pragma unroll
        for (int p = 0; p < 4; ++p) {
            float xr = Ar[col][p], xi = Ai[col][p];
            Ar[col][p] = xr * pr - xi * pi;
            Ai[col][p] = xr * pi + xi * pr;
        }
        { float xr = yr[col], xi = yi[col];
          yr[col] = xr * pr - xi * pi;
          yi[col] = xr * pi + xi * pr; }
        #pragma unroll
        for (int rr = 0; rr < 4; ++rr) {
            if (rr == col) continue;
            float gr = Ar[rr][col], gi = Ai[rr][col];
            #pragma unroll
            for (int p = 0; p < 4; ++p) {
                float xr = Ar[col][p], xi = Ai[col][p];
                Ar[rr][p] -= gr * xr - gi * xi;
                Ai[rr][p] -= gr * xi + gi * xr;
            }
            float xr = yr[col], xi = yi[col];
            yr[rr] -= gr * xr - gi * xi;
            yi[rr] -= gr * xi + gi * xr;
        }
    }

    // res = sum_i unf_save[i] * wiener[i], unf_save = ur + i*ui
    float re = 0.0f, im = 0.0f;
    #pragma unroll
    for (int i = 0; i < 4; ++i) {
        re += ur[i] * yr[i] - ui[i] * yi[i];
        im += ur[i] * yi[i] + ui[i] * yr[i];
    }
    out[((size_t)(b * 2 + 0) * FDIM + f) * TDIM + t] = re;
    out[((size_t)(b * 2 + 1) * FDIM + f) * TDIM + t] = im;
}

extern "C" void kernel_launch(void* const* d_in, const int* in_sizes, int n_in,
                              void* d_out, int out_size, void* d_ws, size_t ws_size,
                              hipStream_t stream) {
    const float* far     = (const float*)d_in[0];
    const float* mix     = (const float*)d_in[1];
    const float* kconv_w = (const float*)d_in[2];
    const float* kconv_b = (const float*)d_in[3];
    const float* ql_w    = (const float*)d_in[4];
    const float* ql_b    = (const float*)d_in[5];
    const float* qn_g    = (const float*)d_in[6];
    const float* qn_b    = (const float*)d_in[7];
    const float* kl_w    = (const float*)d_in[8];
    const float* kl_b    = (const float*)d_in[9];
    const float* kn_g    = (const float*)d_in[10];
    const float* kn_b    = (const float*)d_in[11];
    const float* q_vec   = (const float*)d_in[12];
    const float* k_vec   = (const float*)d_in[13];
    const float* v_vec   = (const float*)d_in[14];
    float* out = (float*)d_out;
    float* W   = (float*)d_ws;        // 16 * 1000 * 16 floats = 1 MB

    // Kernel 1: 64000 GEMM rows / 16 per wave = 4000 single-wave blocks
    wiener_qkw_kernel<<<dim3(4000), dim3(32), 0, stream>>>(
        far, mix, kconv_w, kconv_b, ql_w, ql_b, qn_g, qn_b,
        kl_w, kl_b, kn_g, kn_b, q_vec, k_vec, W);

    // Kernel 2: (b=8) x (f=160) x (4 chunks of 256 t)
    wiener_solve_kernel<<<dim3(8 * FDIM * 4), dim3(256), 0, stream>>>(
        far, mix, v_vec, W, out);
}